// ATTSModel_58067957842619
// MI455X (gfx1250) — compile-verified
//
#include <hip/hip_runtime.h>
#include <hip/hip_bf16.h>
#include <math.h>

typedef __attribute__((ext_vector_type(2))) float v2f;
typedef __attribute__((ext_vector_type(8))) float v8f;

namespace {
constexpr int Tt = 256;   // topic size
constexpr int Ee = 300;   // embedding size
constexpr int Bb = 1024;  // batch
}

// ---------------------------------------------------------------------------
// Kernel 1: M[t,f] = sum_e TE[e,t] * Wf[f,e]    (256x256, K=300)
// One wave per 16x16 tile, V_WMMA_F32_16X16X4_F32, K stepped by 4 (75 steps).
// A layout (16x4 f32): lanes 0-15 hold K={0,1}, lanes 16-31 hold K={2,3}.
// B layout (4x16 f32): mirrored (lane = N, hi half of wave = K+2).
// C/D layout: VGPR r -> row r (lanes 0-15) / row r+8 (lanes 16-31).
// ---------------------------------------------------------------------------
__global__ void k_prep(const float* __restrict__ TE, const float* __restrict__ Wf,
                       float* __restrict__ M) {
  const int lane = threadIdx.x & 31;
  const int wid  = threadIdx.x >> 5;
  const int tile = blockIdx.x * 4 + wid;     // 256 tiles total (16x16 grid)
  const int tm = (tile >> 4) << 4;
  const int tn = (tile & 15) << 4;
  const int m  = lane & 15;
  const int hi = lane >> 4;
  v8f acc = {};
  for (int e0 = 0; e0 < Ee; e0 += 4) {
    const int ka = e0 + hi * 2;
    v2f a, b;
    a.x = TE[(ka + 0) * Tt + tm + m];        // A[m][k] = TE[e][t]
    a.y = TE[(ka + 1) * Tt + tm + m];
    b.x = Wf[(tn + m) * Ee + (ka + 0)];      // B[k][n] = Wf[f][e]
    b.y = Wf[(tn + m) * Ee + (ka + 1)];
    acc = __builtin_amdgcn_wmma_f32_16x16x4_f32(false, a, false, b, (short)0, acc,
                                                false, false);
  }
  const int col = tn + m;
  const int rb  = tm + hi * 8;
#pragma unroll
  for (int r = 0; r < 8; ++r) M[(rb + r) * Tt + col] = acc[r];
}

// ---------------------------------------------------------------------------
// Kernel 2: h[i] = sum_k M[i,k]*b_f[k];  c = sum_k b_f[k]^2
// ---------------------------------------------------------------------------
__global__ void k_hc(const float* __restrict__ M, const float* __restrict__ bf,
                     float* __restrict__ h, float* __restrict__ cOut) {
  __shared__ float s_sq[Tt];
  const int i = threadIdx.x;
  float acc = 0.f;
  for (int k = 0; k < Tt; ++k) acc += M[i * Tt + k] * bf[k];
  h[i] = acc;
  const float b = bf[i];
  s_sq[i] = b * b;
  __syncthreads();
  if (i == 0) {
    float c = 0.f;
    for (int k = 0; k < Tt; ++k) c += s_sq[k];
    cOut[0] = c;
  }
}

// ---------------------------------------------------------------------------
// Kernel 3: G = M * M^T   (256x256, K=256), WMMA f32 16x16x4.  G is symmetric.
// ---------------------------------------------------------------------------
__global__ void k_gh(const float* __restrict__ M, float* __restrict__ G) {
  const int lane = threadIdx.x & 31;
  const int wid  = threadIdx.x >> 5;
  const int tile = blockIdx.x * 4 + wid;
  const int tm = (tile >> 4) << 4;
  const int tn = (tile & 15) << 4;
  const int m  = lane & 15;
  const int hi = lane >> 4;
  v8f acc = {};
  for (int k0 = 0; k0 < Tt; k0 += 4) {
    const int ka = k0 + hi * 2;
    v2f a, b;
    a.x = M[(tm + m) * Tt + ka + 0];         // A[m][k] = M[i][k]
    a.y = M[(tm + m) * Tt + ka + 1];
    b.x = M[(tn + m) * Tt + ka + 0];         // B[k][n] = M[j][k]
    b.y = M[(tn + m) * Tt + ka + 1];
    acc = __builtin_amdgcn_wmma_f32_16x16x4_f32(false, a, false, b, (short)0, acc,
                                                false, false);
  }
  const int col = tn + m;
  const int rb  = tm + hi * 8;
#pragma unroll
  for (int r = 0; r < 8; ++r) G[(rb + r) * Tt + col] = acc[r];
}

// ---------------------------------------------------------------------------
// Kernel 4: per-batch rank-structured att + column softmax + row sums.
//   att[x,y] = rth[x]*wth[y]*G[x,y] + rth[x]*h[x] + wth[y]*h[y] + c
//   softmax over x (axis=1); rs[x0] = sum_y p[x0,y];  u = rs*rth; S = sum rs
// One workgroup (8 waves) per batch; wave handles 32 columns/rows, lanes span
// the reduced dimension (coalesced G row reads thanks to symmetry).
// ---------------------------------------------------------------------------
__global__ void k_att(const float* __restrict__ Rth, const float* __restrict__ Wth,
                      const float* __restrict__ G, const float* __restrict__ h,
                      const float* __restrict__ cPtr,
                      float* __restrict__ U, float* __restrict__ S) {
  __shared__ float s_rth[Tt], s_wth[Tt], s_rh[Tt], s_wc[Tt];
  __shared__ float s_mx[Tt], s_ism[Tt], s_rs[Tt];
  const int b    = blockIdx.x;
  const int tid  = threadIdx.x;
  const int lane = tid & 31;
  const int wid  = tid >> 5;
  const float c  = cPtr[0];
  {
    const float rt = Rth[b * Tt + tid];
    const float wt = Wth[b * Tt + tid];
    const float hv = h[tid];
    s_rth[tid] = rt;
    s_wth[tid] = wt;
    s_rh[tid]  = rt * hv;
    s_wc[tid]  = wt * hv + c;
  }
  __syncthreads();
  // Pass 1: per fixed y, max & expsum over x (softmax normalizers).
  for (int yy = 0; yy < 32; ++yy) {
    const int y = wid * 32 + yy;
    const float wy = s_wth[y], wcy = s_wc[y];
    float av[8];
    float vmax = -3.402823466e+38f;
#pragma unroll
    for (int cc = 0; cc < 8; ++cc) {
      const int x   = cc * 32 + lane;
      const float g = G[y * Tt + x];               // G[x,y] == G[y,x]
      const float a = s_rth[x] * wy * g + s_rh[x] + wcy;
      av[cc] = a;
      vmax   = fmaxf(vmax, a);
    }
#pragma unroll
    for (int o = 16; o > 0; o >>= 1) vmax = fmaxf(vmax, __shfl_xor(vmax, o, 32));
    float ss = 0.f;
#pragma unroll
    for (int cc = 0; cc < 8; ++cc) ss += __expf(av[cc] - vmax);
#pragma unroll
    for (int o = 16; o > 0; o >>= 1) ss += __shfl_xor(ss, o, 32);
    if (lane == 0) { s_mx[y] = vmax; s_ism[y] = 1.f / ss; }
  }
  __syncthreads();
  // Pass 2: rs[x0] = sum_y exp(att[x0,y]-mx[y]) * ism[y]
  for (int xx = 0; xx < 32; ++xx) {
    const int x0 = wid * 32 + xx;
    const float rx = s_rth[x0], rhx = s_rh[x0];
    float acc = 0.f;
#pragma unroll
    for (int cc = 0; cc < 8; ++cc) {
      const int y   = cc * 32 + lane;
      const float g = G[x0 * Tt + y];
      const float a = rx * s_wth[y] * g + rhx + s_wc[y];
      acc += __expf(a - s_mx[y]) * s_ism[y];
    }
#pragma unroll
    for (int o = 16; o > 0; o >>= 1) acc += __shfl_xor(acc, o, 32);
    if (lane == 0) s_rs[x0] = acc;
  }
  __syncthreads();
  U[b * Tt + tid] = s_rs[tid] * s_rth[tid];
  if (wid == 0) {
    float s = 0.f;
#pragma unroll
    for (int cc = 0; cc < 8; ++cc) s += s_rs[cc * 32 + lane];
#pragma unroll
    for (int o = 16; o > 0; o >>= 1) s += __shfl_xor(s, o, 32);
    if (lane == 0) S[b] = s;
  }
}

// ---------------------------------------------------------------------------
// Kernel 5: AR = (1/T)*Rth*M + b_f ;  AW = (1/T)*U*M + (S/T)*b_f
// Two WMMA accumulator chains per wave (shared B operand), 1024x256x256 each.
// ---------------------------------------------------------------------------
__global__ void k_ave(const float* __restrict__ Rth, const float* __restrict__ U,
                      const float* __restrict__ M, const float* __restrict__ bf,
                      const float* __restrict__ S,
                      float* __restrict__ AR, float* __restrict__ AW) {
  const int lane = threadIdx.x & 31;
  const int wid  = threadIdx.x >> 5;
  const int tile = blockIdx.x * 8 + wid;     // 1024 tiles (64 x 16)
  const int tm = (tile >> 4) << 4;           // batch-row base
  const int tn = (tile & 15) << 4;
  const int m  = lane & 15;
  const int hi = lane >> 4;
  v8f accR = {}, accW = {};
  for (int k0 = 0; k0 < Tt; k0 += 4) {
    const int ka = k0 + hi * 2;
    v2f aR, aW, b;
    aR.x = Rth[(tm + m) * Tt + ka + 0];
    aR.y = Rth[(tm + m) * Tt + ka + 1];
    aW.x = U[(tm + m) * Tt + ka + 0];
    aW.y = U[(tm + m) * Tt + ka + 1];
    b.x  = M[(ka + 0) * Tt + tn + m];        // B[k][n] = M[k][n]
    b.y  = M[(ka + 1) * Tt + tn + m];
    accR = __builtin_amdgcn_wmma_f32_16x16x4_f32(false, aR, false, b, (short)0, accR,
                                                 false, false);
    accW = __builtin_amdgcn_wmma_f32_16x16x4_f32(false, aW, false, b, (short)0, accW,
                                                 false, false);
  }
  const int col   = tn + m;
  const float bfc = bf[col];
  const int rb    = tm + hi * 8;
  const float invT = 1.f / (float)Tt;
#pragma unroll
  for (int r = 0; r < 8; ++r) {
    const int row = rb + r;
    AR[row * Tt + col] = accR[r] * invT + bfc;
    AW[row * Tt + col] = accW[r] * invT + S[row] * invT * bfc;
  }
}

// ---------------------------------------------------------------------------
// Kernel 6: cosine similarity * 3, one wave per batch.
// ---------------------------------------------------------------------------
__global__ void k_cos(const float* __restrict__ AR, const float* __restrict__ AW,
                      float* __restrict__ out) {
  const int lane = threadIdx.x & 31;
  const int wid  = threadIdx.x >> 5;
  const int b    = blockIdx.x * 8 + wid;
  float dot = 0.f, n1 = 0.f, n2 = 0.f;
#pragma unroll
  for (int cc = 0; cc < 8; ++cc) {
    const int k   = cc * 32 + lane;
    const float r = AR[b * Tt + k];
    const float w = AW[b * Tt + k];
    dot += r * w;
    n1  += r * r;
    n2  += w * w;
  }
#pragma unroll
  for (int o = 16; o > 0; o >>= 1) {
    dot += __shfl_xor(dot, o, 32);
    n1  += __shfl_xor(n1, o, 32);
    n2  += __shfl_xor(n2, o, 32);
  }
  if (lane == 0) out[b] = 3.f * dot / fmaxf(sqrtf(n1) * sqrtf(n2), 1e-8f);
}

extern "C" void kernel_launch(void* const* d_in, const int* in_sizes, int n_in,
                              void* d_out, int out_size, void* d_ws, size_t ws_size,
                              hipStream_t stream) {
  const float* Rth = (const float*)d_in[0];  // [B,T]
  const float* Wth = (const float*)d_in[1];  // [B,T]
  const float* TE  = (const float*)d_in[2];  // [E,T]
  const float* Wf  = (const float*)d_in[3];  // [T,E]
  const float* bf  = (const float*)d_in[4];  // [T]
  float* out = (float*)d_out;                // [B]

  float* ws = (float*)d_ws;                  // ~3.7 MB of f32 scratch
  float* M  = ws;                            // 65536
  float* G  = M + Tt * Tt;                   // 65536
  float* h  = G + Tt * Tt;                   // 256
  float* c  = h + Tt;                        // 16 (padded)
  float* S  = c + 16;                        // 1024
  float* U  = S + Bb;                        // 262144
  float* AR = U + Bb * Tt;                   // 262144
  float* AW = AR + Bb * Tt;                  // 262144

  k_prep<<<64, 128, 0, stream>>>(TE, Wf, M);
  k_hc<<<1, 256, 0, stream>>>(M, bf, h, c);
  k_gh<<<64, 128, 0, stream>>>(M, G);
  k_att<<<Bb, 256, 0, stream>>>(Rth, Wth, G, h, c, U, S);
  k_ave<<<128, 256, 0, stream>>>(Rth, U, M, bf, S, AR, AW);
  k_cos<<<128, 256, 0, stream>>>(AR, AW, out);
}